// GCN_60301340836134
// MI455X (gfx1250) — compile-verified
//
#include <hip/hip_runtime.h>
#include <hip/hip_bf16.h>

typedef __attribute__((ext_vector_type(2))) float v2f;
typedef __attribute__((ext_vector_type(8))) float v8f;

#define N_NODES 100000
#define N_EDGES 1600000
#define D_IN    128
#define D_OUT   5
#define N_TILES (N_NODES / 16)   // 6250, exact

// ---------------- kernel 0: zero accumulators ----------------
__global__ void GCN_zero_kernel(float* __restrict__ p, int n) {
    int i = blockIdx.x * blockDim.x + threadIdx.x;
    if (i < n) p[i] = 0.0f;
}

// ---------------- kernel 1: fused projection via f32 WMMA ----------------
// y[i, 0:5]  = x[i] @ W_l^T
// y[i, 5:10] = x[i] @ W_r^T
// One wave = one 16-node tile; C/D cols 0..4 -> W_l, 5..9 -> W_r, 10..15 -> 0.
__global__ void GCN_project_wmma(const float* __restrict__ x,
                                 const float* __restrict__ Wl,
                                 const float* __restrict__ Wr,
                                 float* __restrict__ y) {
    const int lane   = threadIdx.x & 31;
    const int wave   = (blockIdx.x * blockDim.x + threadIdx.x) >> 5;
    const int nwaves = (gridDim.x * blockDim.x) >> 5;
    const int half   = lane >> 4;    // 0: K even pair / rows 0..7 ; 1: K odd pair / rows 8..15
    const int mn     = lane & 15;    // M for A-frag, N for B/C/D frags

    // Preload all 32 B fragments (K=0..127 in steps of 4).
    // ISA B layout (mirror of 32-bit A 16x4): lane(n, half) holds B[2*half+v][n]
    // = Wcombined[n][k0 + 2*half + v].
    v2f bq[32];
#pragma unroll
    for (int kk = 0; kk < 32; ++kk) {
        const int k = kk * 4 + 2 * half;
        v2f b;
        if (mn < D_OUT) {
            b = *(const v2f*)(Wl + mn * D_IN + k);
        } else if (mn < 2 * D_OUT) {
            b = *(const v2f*)(Wr + (mn - D_OUT) * D_IN + k);
        } else {
            b = (v2f){0.0f, 0.0f};
        }
        bq[kk] = b;
    }

    for (int tile = wave; tile < N_TILES; tile += nwaves) {
        // A layout: lane(m, half) frag element v holds x[tile*16+m][k0 + 2*half + v]
        const float* xrow = x + (tile * 16 + mn) * D_IN + 2 * half;
        v8f acc = {};
#pragma unroll
        for (int kk = 0; kk < 32; ++kk) {
            v2f a = *(const v2f*)(xrow + kk * 4);   // 8B-aligned global_load_b64
            // (neg_a, A, neg_b, B, c_mod, C, reuse_a, reuse_b)
            acc = __builtin_amdgcn_wmma_f32_16x16x4_f32(
                false, a, false, bq[kk], (short)0, acc, false, false);
        }
        // D layout: VGPR r, lane l -> D[r + 8*(l/16)][l%16]
        if (mn < 2 * D_OUT) {
#pragma unroll
            for (int r = 0; r < 8; ++r) {
                const int row = tile * 16 + r + 8 * half;
                y[row * 10 + mn] = acc[r];
            }
        }
    }
}

// ---------------- kernel 2: edge gather + float-atomic scatter ----------------
__global__ void GCN_edge_scatter(const long long* __restrict__ ei,
                                 const float* __restrict__ y,
                                 float* __restrict__ accum,
                                 float* __restrict__ deg) {
    int e = blockIdx.x * blockDim.x + threadIdx.x;
    if (e >= N_EDGES) return;
    const int src = (int)ei[e];
    const int dst = (int)ei[N_EDGES + e];
    const float* ys = y + src * 10;            // y_l half: 20B, L2-resident table
    v2f y01 = *(const v2f*)(ys);
    v2f y23 = *(const v2f*)(ys + 2);
    float y4 = ys[4];
    float* ad = accum + dst * 5;
    atomicAdd(ad + 0, y01.x);
    atomicAdd(ad + 1, y01.y);
    atomicAdd(ad + 2, y23.x);
    atomicAdd(ad + 3, y23.y);
    atomicAdd(ad + 4, y4);
    atomicAdd(deg + dst, 1.0f);
}

// ---------------- kernel 3: finalize ----------------
__global__ void GCN_finalize(const float* __restrict__ accum,
                             const float* __restrict__ deg,
                             const float* __restrict__ y,
                             const float* __restrict__ bl,
                             float* __restrict__ out) {
    int idx = blockIdx.x * blockDim.x + threadIdx.x;
    if (idx >= N_NODES * D_OUT) return;
    const int i = idx / D_OUT;
    const int c = idx - i * D_OUT;
    float d = deg[i];
    d = d > 1.0f ? d : 1.0f;
    out[idx] = accum[idx] / d + bl[c] + y[i * 10 + D_OUT + c];
}

extern "C" void kernel_launch(void* const* d_in, const int* in_sizes, int n_in,
                              void* d_out, int out_size, void* d_ws, size_t ws_size,
                              hipStream_t stream) {
    const float*     x  = (const float*)d_in[0];
    const long long* ei = (const long long*)d_in[1];   // int64 edge_index [2, E]
    const float*     Wl = (const float*)d_in[2];
    const float*     bl = (const float*)d_in[3];
    const float*     Wr = (const float*)d_in[4];
    float* out = (float*)d_out;

    float* y     = (float*)d_ws;                        // N*10 f32 (4.0 MB)
    float* accum = y + (size_t)N_NODES * 10;            // N*5  f32 (2.0 MB)
    float* deg   = accum + (size_t)N_NODES * D_OUT;     // N    f32 (0.4 MB)
    // accum and deg are contiguous: zero both in one pass.
    const int nzero = N_NODES * (D_OUT + 1);

    GCN_zero_kernel<<<(nzero + 255) / 256, 256, 0, stream>>>(accum, nzero);
    // 256 blocks * 8 waves = 2048 waves grid-striding over 6250 tiles.
    GCN_project_wmma<<<256, 256, 0, stream>>>(x, Wl, Wr, y);
    GCN_edge_scatter<<<(N_EDGES + 255) / 256, 256, 0, stream>>>(ei, y, accum, deg);
    GCN_finalize<<<(N_NODES * D_OUT + 255) / 256, 256, 0, stream>>>(accum, deg, y, bl, out);
}